// GIM_9517647528056
// MI455X (gfx1250) — compile-verified
//
#include <hip/hip_runtime.h>
#include <math.h>

// Shapes: B=4, N=256, D=M=C=128.  Rows = B*N = 1024.
#define ROWS 1024
#define KDIM 128
#define PAD 132   // padded LDS row stride (floats): bank = (4*row + col) % 64 -> conflict free

typedef __attribute__((ext_vector_type(2))) float v2f;
typedef __attribute__((ext_vector_type(8))) float v8f;

// Stage a 128x128 row-major weight matrix into LDS *transposed*: lBt[n*PAD + k] = B[k*128 + n].
// Global side: lanes sweep n for fixed k -> fully coalesced 512B rows.
// LDS side: each thread writes a contiguous k-run (vectorizes to ds_store_b128),
//           banks (4n+k)%64 distinct across lanes -> conflict free.
__device__ __forceinline__ void stage_bt(const float* __restrict__ Bg,
                                         float* __restrict__ lBt, int tid) {
  const int n  = tid & 127;
  const int k0 = (tid >> 7) * 64;   // 256 threads: two k-halves
#pragma unroll 4
  for (int k = k0; k < k0 + 64; k += 4) {
    float x0 = Bg[(k + 0) * KDIM + n];
    float x1 = Bg[(k + 1) * KDIM + n];
    float x2 = Bg[(k + 2) * KDIM + n];
    float x3 = Bg[(k + 3) * KDIM + n];
    float* d = &lBt[n * PAD + k];
    d[0] = x0; d[1] = x1; d[2] = x2; d[3] = x3;
  }
}

// One 16x16 fp32 output tile, K=128, both operands in LDS.
// V_WMMA_F32_16X16X4_F32 operand layout (ISA 7.12.2):
//   A 16x4: lanes 0-15 -> M=lane, K=k0,k0+1; lanes 16-31 -> M=lane-16, K=k0+2,k0+3
//   B 4x16: same split with N=lane%16
//   C/D: VGPR r: lanes 0-15 -> (M=r, N=lane); lanes 16-31 -> (M=r+8, N=lane-16)
__device__ __forceinline__ v8f wmma_k128_lds(const float* __restrict__ lA,
                                             const float* __restrict__ lBt,
                                             int n0, int lane) {
  const int l16  = lane & 15;
  const int koff = (lane >> 4) << 1;    // 0 or 2
  const float* ap = lA  + l16 * PAD + koff;
  const float* bp = lBt + (n0 + l16) * PAD + koff;
  v8f acc = {0.f, 0.f, 0.f, 0.f, 0.f, 0.f, 0.f, 0.f};
#pragma unroll
  for (int k0 = 0; k0 < KDIM; k0 += 4) {
    v2f a, b;
    a.x = ap[k0]; a.y = ap[k0 + 1];     // ds_load_b64 (pairs into ds_load_2addr_b64)
    b.x = bp[k0]; b.y = bp[k0 + 1];     // ds_load_b64
    acc = __builtin_amdgcn_wmma_f32_16x16x4_f32(false, a, false, b,
                                                (short)0, acc, false, false);
  }
  return acc;
}

__device__ __forceinline__ float silu_f(float x) {
  // x * 1/(1+exp(-x)) with raw v_rcp_f32 (1 ulp) instead of the div expansion
  return x * __builtin_amdgcn_rcpf(1.0f + __expf(-x));
}

// Kernel 1: pi = h @ W1a, pj = h @ W1b   (1024x128 @ 128x128, shared A tile, both B in LDS)
__global__ void k_proj(const float* __restrict__ h,
                       const float* __restrict__ W1a,
                       const float* __restrict__ W1b,
                       float* __restrict__ pi, float* __restrict__ pj) {
  __shared__ float lA[16 * PAD];
  __shared__ float lBa[KDIM * PAD];
  __shared__ float lBb[KDIM * PAD];
  const int r0  = blockIdx.x * 16;
  const int tid = threadIdx.x;
  for (int idx = tid; idx < 16 * KDIM; idx += 256)
    lA[(idx >> 7) * PAD + (idx & 127)] = h[(r0 + (idx >> 7)) * KDIM + (idx & 127)];
  stage_bt(W1a, lBa, tid);
  stage_bt(W1b, lBb, tid);
  __syncthreads();

  const int lane = tid & 31, wave = tid >> 5;
  const int n0 = wave * 16, l16 = lane & 15, half = lane >> 4;
  v8f p = wmma_k128_lds(lA, lBa, n0, lane);
  v8f q = wmma_k128_lds(lA, lBb, n0, lane);
#pragma unroll
  for (int r = 0; r < 8; ++r) {
    const int row = r0 + r + half * 8;
    pi[row * KDIM + n0 + l16] = p[r];
    pj[row * KDIM + n0 + l16] = q[r];
  }
}

// Kernel 2 (dominant): per (b,i): S[m] = sum_j adj[b,i,j]*(j!=i)*silu(pi[b,i,m]+pj[b,j,m]+b1[m]),
// A_row = sum_{j!=i} adj, den = max(sum_j adj, 1).
__global__ void k_msg(const float* __restrict__ pi, const float* __restrict__ pj,
                      const float* __restrict__ b1, const float* __restrict__ adj,
                      float* __restrict__ S, float* __restrict__ Arow,
                      float* __restrict__ Den) {
  __shared__ float adjs[256];
  const int bi = blockIdx.x;          // b*256 + i
  const int b  = bi >> 8;
  const int i  = bi & 255;
  const int t  = threadIdx.x;         // m = t, 128 threads

  const float* __restrict__ ar = adj + (size_t)bi * 256;
  for (int j = t; j < 256; j += 128) adjs[j] = ar[j];
  __syncthreads();

  const float base = pi[bi * KDIM + t] + b1[t];
  const float* __restrict__ pjb = pj + (size_t)b * 256 * KDIM + t;

  float acc = 0.f, rs = 0.f;
#pragma unroll 4
  for (int j = 0; j < 256; ++j) {
    const float w = adjs[j];
    rs += w;
    const float x = base + pjb[j * KDIM];
    const float s = silu_f(x);
    const float we = (j == i) ? 0.f : w;  // diagonal mask
    acc = fmaf(we, s, acc);
  }
  S[bi * KDIM + t] = acc;
  if (t == 0) {
    Arow[bi] = rs - adjs[i];
    Den[bi]  = fmaxf(rs, 1.0f);
  }
}

// Kernel 3: m_agg = (S @ W2 + b2 * A_row) / den
__global__ void k_agg(const float* __restrict__ S, const float* __restrict__ W2,
                      const float* __restrict__ b2, const float* __restrict__ Arow,
                      const float* __restrict__ Den, float* __restrict__ magg) {
  __shared__ float lA[16 * PAD];
  __shared__ float lB[KDIM * PAD];
  const int r0  = blockIdx.x * 16;
  const int tid = threadIdx.x;
  for (int idx = tid; idx < 16 * KDIM; idx += 256)
    lA[(idx >> 7) * PAD + (idx & 127)] = S[(r0 + (idx >> 7)) * KDIM + (idx & 127)];
  stage_bt(W2, lB, tid);
  __syncthreads();

  const int lane = tid & 31, wave = tid >> 5;
  const int n0 = wave * 16, l16 = lane & 15, half = lane >> 4;
  v8f acc = wmma_k128_lds(lA, lB, n0, lane);
  const float bcol = b2[n0 + l16];
#pragma unroll
  for (int r = 0; r < 8; ++r) {
    const int row = r0 + r + half * 8;
    const float rd = __builtin_amdgcn_rcpf(Den[row]);
    magg[row * KDIM + n0 + l16] = (acc[r] + bcol * Arow[row]) * rd;
  }
}

// Kernel 4: out = silu(m_agg @ Wc1 + bc1) @ Wc2 + bc2   (two GEMMs fused through LDS)
__global__ void k_head(const float* __restrict__ magg,
                       const float* __restrict__ Wc1, const float* __restrict__ bc1,
                       const float* __restrict__ Wc2, const float* __restrict__ bc2,
                       float* __restrict__ out) {
  __shared__ float lA[16 * PAD];
  __shared__ float lT[16 * PAD];
  __shared__ float lB1[KDIM * PAD];
  __shared__ float lB2[KDIM * PAD];
  const int r0  = blockIdx.x * 16;
  const int tid = threadIdx.x;
  for (int idx = tid; idx < 16 * KDIM; idx += 256)
    lA[(idx >> 7) * PAD + (idx & 127)] = magg[(r0 + (idx >> 7)) * KDIM + (idx & 127)];
  stage_bt(Wc1, lB1, tid);
  stage_bt(Wc2, lB2, tid);
  __syncthreads();

  const int lane = tid & 31, wave = tid >> 5;
  const int n0 = wave * 16, l16 = lane & 15, half = lane >> 4;

  v8f u = wmma_k128_lds(lA, lB1, n0, lane);
  const float b1c = bc1[n0 + l16];
#pragma unroll
  for (int r = 0; r < 8; ++r) {
    const int rr = r + half * 8;
    lT[rr * PAD + n0 + l16] = silu_f(u[r] + b1c);
  }
  __syncthreads();

  v8f c = wmma_k128_lds(lT, lB2, n0, lane);
  const float b2c = bc2[n0 + l16];
#pragma unroll
  for (int r = 0; r < 8; ++r) {
    const int row = r0 + r + half * 8;
    out[row * KDIM + n0 + l16] = c[r] + b2c;
  }
}

extern "C" void kernel_launch(void* const* d_in, const int* in_sizes, int n_in,
                              void* d_out, int out_size, void* d_ws, size_t ws_size,
                              hipStream_t stream) {
  const float* h   = (const float*)d_in[0];
  const float* adj = (const float*)d_in[1];
  const float* W1a = (const float*)d_in[2];
  const float* W1b = (const float*)d_in[3];
  const float* b1  = (const float*)d_in[4];
  const float* W2  = (const float*)d_in[5];
  const float* b2  = (const float*)d_in[6];
  const float* Wc1 = (const float*)d_in[7];
  const float* bc1 = (const float*)d_in[8];
  const float* Wc2 = (const float*)d_in[9];
  const float* bc2 = (const float*)d_in[10];
  float* out = (float*)d_out;

  float* ws = (float*)d_ws;
  float* pi = ws;                       // 1024*128
  float* pj = ws + 131072;              // 1024*128
  float* S  = ws + 262144;              // 1024*128
  float* Ar = ws + 393216;              // 1024
  float* Dn = ws + 394240;              // 1024
  float* mg = ws + 395264;              // 1024*128

  k_proj<<<ROWS / 16, 256, 0, stream>>>(h, W1a, W1b, pi, pj);
  k_msg <<<ROWS, 128, 0, stream>>>(pi, pj, b1, adj, S, Ar, Dn);
  k_agg <<<ROWS / 16, 256, 0, stream>>>(S, W2, b2, Ar, Dn, mg);
  k_head<<<ROWS / 16, 256, 0, stream>>>(mg, Wc1, bc1, Wc2, bc2, out);
}